// GraphSAGEWithEdgeFeatures_22368189678239
// MI455X (gfx1250) — compile-verified
//
#include <hip/hip_runtime.h>

// f32 WMMA fragments: A 16x4 -> 2 VGPRs/lane, B 4x16 -> 2 VGPRs/lane, C/D 16x16 -> 8 VGPRs/lane
typedef float v2f __attribute__((ext_vector_type(2)));
typedef float v8f __attribute__((ext_vector_type(8)));

#define WAVES   2      // waves per block
#define MT      2      // 16-edge M-tiles per wave (B-fragment reuse factor)
#define H1_LD   132    // LDS row stride (floats): 128 + 4 pad -> conflict-free b64 reads

__global__ __launch_bounds__(256) void zero_out_kernel(float* __restrict__ out, int n) {
  int i = blockIdx.x * blockDim.x + threadIdx.x;
  if (i < n) out[i] = 0.0f;
}

__global__ __launch_bounds__(WAVES * 32) void edge_mlp_kernel(
    const float* __restrict__ x,          // [4096,128]
    const long long* __restrict__ row,    // [E]
    const long long* __restrict__ col,    // [E]
    const float* __restrict__ ea,         // [4096,4096,16]
    const float* __restrict__ W1,         // [144,128]
    const float* __restrict__ b1,         // [128]
    const float* __restrict__ W2,         // [128,128]
    const float* __restrict__ b2,         // [128]
    float* __restrict__ out)              // [4096,128]
{
  __shared__ float h1s[WAVES][MT * 16 * H1_LD];

  const int tid  = threadIdx.x;
  const int wave = tid >> 5;
  const int lane = tid & 31;
  const int m    = lane & 15;   // A-matrix row / C-matrix column within a 16-tile
  const int kh   = lane >> 4;   // K half: lanes 16-31 hold K+2,K+3 of each 4-step
  const int te   = ((int)blockIdx.x * WAVES + wave) * (MT * 16); // first edge of this wave

  // per-lane gather bases for the two A tiles (edges te+m and te+16+m)
  const long long r0 = row[te + m];
  const long long c0 = col[te + m];
  const long long r1 = row[te + 16 + m];
  const long long c1 = col[te + 16 + m];
  const float* __restrict__ xrow0  = x  + (size_t)r0 * 128;
  const float* __restrict__ xrow1  = x  + (size_t)r1 * 128;
  const float* __restrict__ earow0 = ea + (((size_t)r0 * 4096 + (size_t)c0) * 16);
  const float* __restrict__ earow1 = ea + (((size_t)r1 * 4096 + (size_t)c1) * 16);

  // ---------------- layer 1: H1 = relu([x_row | e] @ W1 + b1), K = 144 ----------------
  v8f acc[MT][8];
#pragma unroll
  for (int n = 0; n < 8; ++n) {
    float bv = b1[n * 16 + m];                  // C/D layout: lane holds column n*16 + m
    acc[0][n] = (v8f){bv, bv, bv, bv, bv, bv, bv, bv};
    acc[1][n] = acc[0][n];
  }

  // K = 0..127 from gathered x rows (L2-resident)
  for (int ks = 0; ks < 32; ++ks) {
    const int kk = ks * 4 + kh * 2;
    v2f a0 = *(const v2f*)(xrow0 + kk);         // one b64 per lane per tile
    v2f a1 = *(const v2f*)(xrow1 + kk);
#pragma unroll
    for (int n = 0; n < 8; ++n) {
      const int cidx = n * 16 + m;
      v2f b;                                    // B fragment loaded ONCE, feeds 2 WMMAs
      b.x = W1[(size_t)kk * 128 + cidx];
      b.y = W1[(size_t)(kk + 1) * 128 + cidx];
      acc[0][n] = __builtin_amdgcn_wmma_f32_16x16x4_f32(false, a0, false, b,
                                                        (short)0, acc[0][n], false, false);
      acc[1][n] = __builtin_amdgcn_wmma_f32_16x16x4_f32(false, a1, false, b,
                                                        (short)0, acc[1][n], false, false);
    }
  }
  // K = 128..143 from gathered edge_attr rows (aligned 64B lines)
  for (int ks = 0; ks < 4; ++ks) {
    const int kk = ks * 4 + kh * 2;
    v2f a0 = *(const v2f*)(earow0 + kk);
    v2f a1 = *(const v2f*)(earow1 + kk);
    const int kw = 128 + kk;
#pragma unroll
    for (int n = 0; n < 8; ++n) {
      const int cidx = n * 16 + m;
      v2f b;
      b.x = W1[(size_t)kw * 128 + cidx];
      b.y = W1[(size_t)(kw + 1) * 128 + cidx];
      acc[0][n] = __builtin_amdgcn_wmma_f32_16x16x4_f32(false, a0, false, b,
                                                        (short)0, acc[0][n], false, false);
      acc[1][n] = __builtin_amdgcn_wmma_f32_16x16x4_f32(false, a1, false, b,
                                                        (short)0, acc[1][n], false, false);
    }
  }

  // relu + stash H1 (32 edges x 128 ch) into LDS, C/D layout -> padded row-major
  float* __restrict__ hs = h1s[wave];
#pragma unroll
  for (int t = 0; t < MT; ++t) {
#pragma unroll
    for (int n = 0; n < 8; ++n) {
#pragma unroll
      for (int r = 0; r < 8; ++r) {
        hs[(t * 16 + r + 8 * kh) * H1_LD + n * 16 + m] = fmaxf(acc[t][n][r], 0.0f);
      }
    }
  }

  // ---------------- layer 2: H2 = relu(H1 @ W2 + b2), K = 128 ----------------
  v8f acc2[MT][8];
#pragma unroll
  for (int n = 0; n < 8; ++n) {
    float bv = b2[n * 16 + m];
    acc2[0][n] = (v8f){bv, bv, bv, bv, bv, bv, bv, bv};
    acc2[1][n] = acc2[0][n];
  }

  for (int ks = 0; ks < 32; ++ks) {
    const int kk = ks * 4 + kh * 2;
    v2f a0 = *(const v2f*)(hs + (m)      * H1_LD + kk);  // ds_load_b64, conflict-free
    v2f a1 = *(const v2f*)(hs + (m + 16) * H1_LD + kk);
#pragma unroll
    for (int n = 0; n < 8; ++n) {
      const int cidx = n * 16 + m;
      v2f b;
      b.x = W2[(size_t)kk * 128 + cidx];
      b.y = W2[(size_t)(kk + 1) * 128 + cidx];
      acc2[0][n] = __builtin_amdgcn_wmma_f32_16x16x4_f32(false, a0, false, b,
                                                         (short)0, acc2[0][n], false, false);
      acc2[1][n] = __builtin_amdgcn_wmma_f32_16x16x4_f32(false, a1, false, b,
                                                         (short)0, acc2[1][n], false, false);
    }
  }

  // ---------------- relu + scatter-add into out[col] (hw f32 atomics, L2-hot) ----------------
#pragma unroll
  for (int t = 0; t < MT; ++t) {
#pragma unroll
    for (int r = 0; r < 8; ++r) {
      const long long cdst = col[te + t * 16 + r + 8 * kh]; // edge M = t*16 + r + 8*kh
      float* __restrict__ obase = out + (size_t)cdst * 128;
#pragma unroll
      for (int n = 0; n < 8; ++n) {
        float v = fmaxf(acc2[t][n][r], 0.0f);
        __hip_atomic_fetch_add(obase + n * 16 + m, v,
                               __ATOMIC_RELAXED, __HIP_MEMORY_SCOPE_AGENT);
      }
    }
  }
}

extern "C" void kernel_launch(void* const* d_in, const int* in_sizes, int n_in,
                              void* d_out, int out_size, void* d_ws, size_t ws_size,
                              hipStream_t stream) {
  const float*     x  = (const float*)d_in[0];
  const long long* ei = (const long long*)d_in[1];   // int64 [2, E]
  const float*     ea = (const float*)d_in[2];
  const float*     W1 = (const float*)d_in[3];
  const float*     b1 = (const float*)d_in[4];
  const float*     W2 = (const float*)d_in[5];
  const float*     b2 = (const float*)d_in[6];
  float* out = (float*)d_out;

  const int E = in_sizes[1] / 2;

  // out is poisoned by the harness; atomics need a zero base
  zero_out_kernel<<<(out_size + 255) / 256, 256, 0, stream>>>(out, out_size);

  const int edges_per_block = WAVES * MT * 16;       // 64
  edge_mlp_kernel<<<E / edges_per_block, WAVES * 32, 0, stream>>>(
      x, ei, ei + E, ea, W1, b1, W2, b2, out);
}